// FeCAMClassifier_51488067944885
// MI455X (gfx1250) — compile-verified
//
#include <hip/hip_runtime.h>

// FeCAM logits on MI455X (gfx1250, wave32).
// logits[n,c] = -0.5*|| L[c]^{-1} (x[n]-mu[c])/std[c] ||^2 - 0.5*logdetR[c]
//
// Blocked forward substitution, one workgroup per (class, 64-point tile).
// Off-diagonal updates use V_WMMA_F32_16X16X4_F32 (f32 kept: the triangular
// solve feeds errors forward through 48 dependent steps; bf16 unacceptable).
// Full Y panel (768x64 f32) resident in the 320KB WGP LDS.
//
// v3: split-K loop bounds go through readfirstlane so the compiler emits
// scalar branches (no exec masking around WMMA); L-panel staging uses
// float4 (global_load_b128 / ds_store_b128) instead of scalar b32.

#define NPTS   1024
#define NCLS   200
#define DIM    768
#define FEPS   1e-6f

#define TILE_N 64          // RHS columns per workgroup
#define NBLK   (DIM / 16)  // 48 diagonal blocks
#define NTHR   256         // 8 waves

#define LSTR   772         // L row-panel LDS stride (772%64==4 -> conflict-free A reads)
#define YSTR   72          // Y LDS stride (72%64==8 -> half-waves hit disjoint banks)
#define ASTR   17          // acc spill stride

typedef __attribute__((ext_vector_type(2))) float v2f;
typedef __attribute__((ext_vector_type(8))) float v8f;

extern "C" __global__ __launch_bounds__(NTHR)
void fecam_solve_kernel(const float* __restrict__ X,
                        const float* __restrict__ MU,
                        const float* __restrict__ SD,
                        const float* __restrict__ L,
                        const float* __restrict__ LOGDET,
                        float* __restrict__ OUT)
{
    extern __shared__ float lds[];
    float* Llds = lds;                        // 16  * LSTR floats (L row panel)
    float* Ylds = lds + 16 * LSTR;            // DIM * YSTR floats (solved Y)
    float* Alds = Ylds + DIM * YSTR;          // 8 * 16 * ASTR floats (acc spills)

    const int c    = blockIdx.x / (NPTS / TILE_N);
    const int nt   = blockIdx.x % (NPTS / TILE_N);
    const int n0   = nt * TILE_N;
    const int tid  = threadIdx.x;
    const int wave = tid >> 5;
    const int lane = tid & 31;
    const int lm   = lane & 15;
    const int hi   = lane >> 4;      // half-wave (ISA f32 fragment layout)

    // wave-uniform values -> force into SGPRs so loop bounds compile to
    // scalar branches (keeps EXEC all-1s through the WMMA loop)
    const int kpar = __builtin_amdgcn_readfirstlane(wave >> 2);   // split-K parity
    const int nw   = __builtin_amdgcn_readfirstlane((wave & 3) * 16); // column slice

    const float* Lc = L + (size_t)c * DIM * DIM;

    float dsum = 0.0f;               // per solver thread (tid<64): ||y||^2 of column tid

    // load one jb's worth of fragments (K=16 -> 4 WMMA K-chunks)
    auto loadfrag = [&](int jb, v2f* a, v2f* b) {
        const int col0 = jb * 16;
        #pragma unroll
        for (int k = 0; k < 4; ++k) {
            const int ka = col0 + 4 * k + 2 * hi;
            a[k].x = Llds[lm * LSTR + ka];
            a[k].y = Llds[lm * LSTR + ka + 1];
            b[k].x = Ylds[(ka    ) * YSTR + nw + lm];
            b[k].y = Ylds[(ka + 1) * YSTR + nw + lm];
        }
    };

    for (int kb = 0; kb < NBLK; ++kb) {
        const int r0    = kb * 16;
        const int ncols = r0 + 16;   // need columns [0, r0+16): off-diag + diag block

        // ---- stage L row-panel rows [r0, r0+16) x cols [0, ncols) into LDS ----
        // 16 threads per row, float4 -> global_load_b128 / ds_store_b128
        {
            const int r = tid >> 4;
            const int t = tid & 15;
            const float* src = Lc + (size_t)(r0 + r) * DIM;
            for (int i = t; i < (ncols >> 2); i += 16)
                *(float4*)&Llds[r * LSTR + 4 * i] = ((const float4*)src)[i];
        }
        // hint next panel toward the caches (global_prefetch_b8), one row per
        // 16-thread group, strided across the row
        if (kb + 1 < NBLK)
            __builtin_prefetch(Lc + (size_t)(r0 + 16 + (tid >> 4)) * DIM + (tid & 15) * 48,
                               0, 1);
        __syncthreads();

        // ---- acc(16x16) += L[kb, jb] @ Y[jb, nw:nw+16], jb = kpar, kpar+2, ... ----
        // Double-buffered fragments: load jb+2 while the WMMAs for jb issue.
        v8f acc = 0.0f;
        {
            v2f aA[4], bA[4], aB[4], bB[4];
            int jb = kpar;
            if (jb < kb) loadfrag(jb, aA, bA);
            while (jb < kb) {
                int jn = jb + 2;
                if (jn < kb) loadfrag(jn, aB, bB);
                #pragma unroll
                for (int k = 0; k < 4; ++k)
                    acc = __builtin_amdgcn_wmma_f32_16x16x4_f32(
                            false, aA[k], false, bA[k], (short)0, acc, false, false);
                jb = jn;
                if (jb >= kb) break;
                jn = jb + 2;
                if (jn < kb) loadfrag(jn, aA, bA);
                #pragma unroll
                for (int k = 0; k < 4; ++k)
                    acc = __builtin_amdgcn_wmma_f32_16x16x4_f32(
                            false, aB[k], false, bB[k], (short)0, acc, false, false);
                jb = jn;
            }
        }

        // spill fragment: VGPR r, half-wave hi -> element (M = r + 8*hi, N = lm)
        #pragma unroll
        for (int r = 0; r < 8; ++r)
            Alds[wave * 16 * ASTR + (r + 8 * hi) * ASTR + lm] = acc[r];
        __syncthreads();

        // ---- diagonal 16x16 triangular solve; 64 threads = 64 RHS columns ----
        if (tid < TILE_N) {
            const int    nG = n0 + tid;
            const float* xp = X + (size_t)nG * DIM + r0;
            const int    aw = tid >> 4;     // producing tile index
            const int    an = tid & 15;     // column within that tile
            float y[16];
            #pragma unroll
            for (int m = 0; m < 16; ++m) {
                const int   d  = r0 + m;
                const float sd = SD[c * DIM + d];
                const float w  = (xp[m] - MU[c * DIM + d]) / fmaxf(sd, FEPS);
                // sum the two split-K partial accumulators (waves aw and aw+4)
                float v = w - Alds[aw * 16 * ASTR + m * ASTR + an]
                            - Alds[(aw + 4) * 16 * ASTR + m * ASTR + an];
                for (int j = 0; j < m; ++j)
                    v -= Llds[m * LSTR + r0 + j] * y[j];
                v /= Llds[m * LSTR + r0 + m];
                y[m] = v;
                dsum += v * v;
                Ylds[d * YSTR + tid] = v;   // publish for future WMMA B-frags
            }
        }
        __syncthreads();
    }

    if (tid < TILE_N) {
        const int nG = n0 + tid;
        OUT[(size_t)nG * NCLS + c] = -0.5f * dsum - 0.5f * LOGDET[c];
    }
}

extern "C" void kernel_launch(void* const* d_in, const int* in_sizes, int n_in,
                              void* d_out, int out_size, void* d_ws, size_t ws_size,
                              hipStream_t stream) {
    (void)in_sizes; (void)n_in; (void)out_size; (void)d_ws; (void)ws_size;
    const float* X      = (const float*)d_in[0];   // (1024, 768)
    const float* MU     = (const float*)d_in[1];   // (200, 768)
    const float* SD     = (const float*)d_in[2];   // (200, 768)
    const float* L      = (const float*)d_in[3];   // (200, 768, 768)
    const float* LOGDET = (const float*)d_in[4];   // (200,)
    float* OUT = (float*)d_out;                    // (1024, 200)

    const size_t shmem = (size_t)(16 * LSTR + DIM * YSTR + 8 * 16 * ASTR) * sizeof(float);
    dim3 grid(NCLS * (NPTS / TILE_N));             // 3200 workgroups
    fecam_solve_kernel<<<grid, NTHR, shmem, stream>>>(X, MU, SD, L, LOGDET, OUT);
}